// Net_6322191859870
// MI455X (gfx1250) — compile-verified
//
#include <hip/hip_runtime.h>
#include <math.h>

#define TAU 0.5f
#define NEG_SLOPE 0.1f
#define N_USER 100000
#define N_MOVIE 50000
#define R_REL 5
#define E_EDGE 100000
#define F_DIM 2
#define D_DIM 32
#define RD_DIM 64
#define OUT_DIM 64

typedef __attribute__((ext_vector_type(2))) float v2f;
typedef __attribute__((ext_vector_type(8))) float v8f;

// ---------------------------------------------------------------------------
// Zero accumulator regions of workspace (grid-stride, b128 stores).
// ---------------------------------------------------------------------------
__global__ void zero_kernel(float4* __restrict__ p, long n4) {
    long i = (long)blockIdx.x * blockDim.x + threadIdx.x;
    long stride = (long)gridDim.x * blockDim.x;
    float4 z = make_float4(0.f, 0.f, 0.f, 0.f);
    for (; i < n4; i += stride) p[i] = z;
}

// ---------------------------------------------------------------------------
// fp32 WMMA GEMM: C[r] = act(A[r]) @ B[r] (+ bias).
// One wave owns a 16-row M-tile and ALL NT column tiles (NT = N/16 <= 4),
// so each A fragment is loaded exactly once per K-step and feeds NT
// independent V_WMMA_F32_16X16X4_F32 ops (co-executable, no RAW chain
// between them). M is a multiple of 16 for every use; tail waves exit
// uniformly so EXEC is all-1s inside WMMA (ISA 7.12 requirement).
//
// Layouts per ISA 7.12.2:
//   A 16x4 f32 : lane L -> row (L&15), VGPR j -> K = (L>>4)*2 + j
//   B 4x16 f32 : lane L -> col (L&15), VGPR j -> K = (L>>4)*2 + j
//   C/D 16x16  : VGPR v -> row v (lanes 0-15) / v+8 (lanes 16-31), col L&15
// ---------------------------------------------------------------------------
template <int LEAKY, int NT>
__global__ void wmma_gemm(const float* __restrict__ Aall,
                          const float* __restrict__ Ball,
                          const float* __restrict__ bias,
                          float* __restrict__ Call,
                          int M, int Kdim,
                          long strideA, long strideB, long strideC) {
    const int N = NT * 16;
    const int wave = threadIdx.x >> 5;
    const int lane = threadIdx.x & 31;
    const int mt   = blockIdx.x * 8 + wave;
    const int numMt = M >> 4;
    if (mt >= numMt) return;               // wave-uniform guard
    const int r = blockIdx.z;

    const float* A = Aall + (long)r * strideA;
    const float* B = Ball + (long)r * strideB;
    float*       C = Call + (long)r * strideC;

    const int mrow = (mt << 4) + (lane & 15);
    const int kh   = (lane >> 4) << 1;     // 0 or 2
    const int nlo  = lane & 15;

    v8f acc[NT];
#pragma unroll
    for (int t = 0; t < NT; ++t) acc[t] = (v8f){};

    for (int kk = 0; kk < Kdim; kk += 4) {
        v2f a;
        const float* ap = A + (long)mrow * Kdim + kk + kh;
        a.x = ap[0];
        a.y = ap[1];
        if (LEAKY) {
            a.x = a.x >= 0.0f ? a.x : NEG_SLOPE * a.x;
            a.y = a.y >= 0.0f ? a.y : NEG_SLOPE * a.y;
        }
        const float* bp0 = B + (long)(kk + kh) * N + nlo;
        const float* bp1 = B + (long)(kk + kh + 1) * N + nlo;
#pragma unroll
        for (int t = 0; t < NT; ++t) {
            v2f b;
            b.x = bp0[t * 16];
            b.y = bp1[t * 16];
            acc[t] = __builtin_amdgcn_wmma_f32_16x16x4_f32(
                         false, a, false, b, (short)0, acc[t], false, false);
        }
    }

    const int rbase = (mt << 4) + ((lane >> 4) << 3);
#pragma unroll
    for (int t = 0; t < NT; ++t) {
        const int ncol = t * 16 + nlo;
        const float bv = bias ? bias[ncol] : 0.0f;
#pragma unroll
        for (int v = 0; v < 8; ++v) {
            C[(long)(rbase + v) * N + ncol] = acc[t][v] + bv;
        }
    }
}

// ---------------------------------------------------------------------------
// Per-edge weight computation + atomic degree sums.
// grid: (ceil(E/256), R), block 256. One thread per edge; all row reads are
// 128B-aligned float4 (global_load_b128).
// ---------------------------------------------------------------------------
__global__ void edge_weights(const float* __restrict__ fku,
                             const float* __restrict__ fkm,
                             const float* __restrict__ fsu,
                             const float* __restrict__ fsm,
                             const float* __restrict__ rfa,
                             const float* __restrict__ proto,
                             const float* __restrict__ eta,
                             const int*   __restrict__ erow,
                             const int*   __restrict__ ecol,
                             float* __restrict__ w_raw,
                             float* __restrict__ norm_user,
                             float* __restrict__ norm_movie) {
    __shared__ float4 sproto[F_DIM * RD_DIM / 4];
    for (int i = threadIdx.x; i < F_DIM * RD_DIM / 4; i += blockDim.x)
        sproto[i] = ((const float4*)proto)[i];
    __syncthreads();

    const int r = blockIdx.y;
    const int e = blockIdx.x * blockDim.x + threadIdx.x;
    if (e >= E_EDGE) return;
    const long t = (long)r * E_EDGE + e;
    const int row = erow[t];
    const int col = ecol[t];

    // cosine similarity with clipped norms
    const float4* ku = (const float4*)(fku + ((long)r * N_USER + row) * D_DIM);
    const float4* km = (const float4*)(fkm + ((long)r * N_MOVIE + col) * D_DIM);
    float nu = 0.f, nm = 0.f, dt = 0.f;
#pragma unroll
    for (int d = 0; d < D_DIM / 4; ++d) {
        float4 a = ku[d], b = km[d];
        nu += a.x * a.x + a.y * a.y + a.z * a.z + a.w * a.w;
        nm += b.x * b.x + b.y * b.y + b.z * b.z + b.w * b.w;
        dt += a.x * b.x + a.y * b.y + a.z * b.z + a.w * b.w;
    }
    nu = fmaxf(sqrtf(nu), 1e-12f);
    nm = fmaxf(sqrtf(nm), 1e-12f);
    const float sim_k = dt / (nu * nm) / TAU;

    // summary similarities (F=2)
    const float4* su = (const float4*)(fsu + ((long)r * N_USER + row) * F_DIM * D_DIM);
    const float4* sm = (const float4*)(fsm + ((long)r * N_MOVIE + col) * F_DIM * D_DIM);
    float s0 = 0.f, s1 = 0.f;
#pragma unroll
    for (int d = 0; d < D_DIM / 4; ++d) {
        float4 a = su[d], b = sm[d];
        s0 += a.x * b.x + a.y * b.y + a.z * b.z + a.w * b.w;
        float4 c = su[D_DIM / 4 + d], g2 = sm[D_DIM / 4 + d];
        s1 += c.x * g2.x + c.y * g2.y + c.z * g2.z + c.w * g2.w;
    }
    const float exp_sim = expf(sim_k) / (expf(s0 / TAU) + expf(s1 / TAU));

    // anchor similarities (K = 0); prefetch next block's review rows
    const float* rvf = rfa + ((long)r * E_EDGE + e) * F_DIM * RD_DIM;
    __builtin_prefetch(rvf + (long)blockDim.x * F_DIM * RD_DIM, 0, 1);
    const float4* rv = (const float4*)rvf;
    float a0 = 0.f, a1 = 0.f;
#pragma unroll
    for (int k = 0; k < RD_DIM / 4; ++k) {
        float4 a = rv[k], p = sproto[k];
        a0 += a.x * p.x + a.y * p.y + a.z * p.z + a.w * p.w;
        float4 b = rv[RD_DIM / 4 + k], q = sproto[RD_DIM / 4 + k];
        a1 += b.x * q.x + b.y * q.y + b.z * q.z + b.w * q.w;
    }
    const float ea0 = expf(a0 / TAU);
    const float exp_anchor = ea0 / (ea0 + expf(a1 / TAU));

    const float g = 1.0f / (1.0f + expf(-eta[t]));
    const float w = g * exp_anchor + (1.0f - g) * exp_sim;

    w_raw[t] = w;
    atomicAdd(norm_user + row, w);    // (seg(wf)+seg(wr))*0.5 == seg(w_raw)
    atomicAdd(norm_movie + col, w);
}

// ---------------------------------------------------------------------------
// Per-edge messages: normalize weight, write int_dist, fused review matvec
// (weights staged in LDS, broadcast reads), atomic scatter into node accums.
// grid: (ceil(E/256), R), block 256.
// ---------------------------------------------------------------------------
__global__ void edge_messages(const float* __restrict__ src_u,
                              const float* __restrict__ src_m,
                              const float* __restrict__ rfa,
                              const float* __restrict__ rwf,
                              const float* __restrict__ rwr,
                              const int*   __restrict__ erow,
                              const int*   __restrict__ ecol,
                              const float* __restrict__ w_raw,
                              const float* __restrict__ norm_user,
                              const float* __restrict__ norm_movie,
                              float* __restrict__ ufeat_pre,
                              float* __restrict__ ifeat_pre,
                              float* __restrict__ int_dist) {
    __shared__ float swf[RD_DIM * D_DIM];   // 8 KB
    __shared__ float swr[RD_DIM * D_DIM];   // 8 KB
    const int r = blockIdx.y;
    for (int i = threadIdx.x; i < RD_DIM * D_DIM / 4; i += blockDim.x) {
        ((float4*)swf)[i] = ((const float4*)(rwf + (long)r * RD_DIM * D_DIM))[i];
        ((float4*)swr)[i] = ((const float4*)(rwr + (long)r * RD_DIM * D_DIM))[i];
    }
    __syncthreads();

    const int e = blockIdx.x * blockDim.x + threadIdx.x;
    if (e >= E_EDGE) return;
    const long t = (long)r * E_EDGE + e;
    const int row = erow[t];
    const int col = ecol[t];

    const float w   = w_raw[t];
    const float nij = sqrtf(norm_user[row] * norm_movie[col]);
    const float wn  = w / nij;
    int_dist[t] = wn;

    const float4* rv = (const float4*)(rfa + ((long)r * E_EDGE + e) * F_DIM * RD_DIM);
    float acc[D_DIM];

    // forward message -> movie accumulator
    const float4* su = (const float4*)(src_u + ((long)r * N_USER + row) * D_DIM);
#pragma unroll
    for (int d = 0; d < D_DIM / 4; ++d) {
        float4 v = su[d];
        acc[4 * d] = v.x; acc[4 * d + 1] = v.y; acc[4 * d + 2] = v.z; acc[4 * d + 3] = v.w;
    }
    for (int k4 = 0; k4 < RD_DIM / 4; ++k4) {
        float4 rk = rv[k4];
        const float* wrow = swf + 4 * k4 * D_DIM;
#pragma unroll
        for (int d = 0; d < D_DIM; ++d)
            acc[d] += rk.x * wrow[d] + rk.y * wrow[D_DIM + d]
                    + rk.z * wrow[2 * D_DIM + d] + rk.w * wrow[3 * D_DIM + d];
    }
    {
        float* op = ifeat_pre + (long)col * D_DIM;
#pragma unroll
        for (int d = 0; d < D_DIM; ++d) atomicAdd(op + d, acc[d] * wn);
    }

    // reverse message -> user accumulator
    const float4* sm2 = (const float4*)(src_m + ((long)r * N_MOVIE + col) * D_DIM);
#pragma unroll
    for (int d = 0; d < D_DIM / 4; ++d) {
        float4 v = sm2[d];
        acc[4 * d] = v.x; acc[4 * d + 1] = v.y; acc[4 * d + 2] = v.z; acc[4 * d + 3] = v.w;
    }
    for (int k4 = 0; k4 < RD_DIM / 4; ++k4) {
        float4 rk = rv[k4];
        const float* wrow = swr + 4 * k4 * D_DIM;
#pragma unroll
        for (int d = 0; d < D_DIM; ++d)
            acc[d] += rk.x * wrow[d] + rk.y * wrow[D_DIM + d]
                    + rk.z * wrow[2 * D_DIM + d] + rk.w * wrow[3 * D_DIM + d];
    }
    {
        float* op = ufeat_pre + (long)row * D_DIM;
#pragma unroll
        for (int d = 0; d < D_DIM; ++d) atomicAdd(op + d, acc[d] * wn);
    }
}

// ---------------------------------------------------------------------------
extern "C" void kernel_launch(void* const* d_in, const int* in_sizes, int n_in,
                              void* d_out, int out_size, void* d_ws, size_t ws_size,
                              hipStream_t stream) {
    const float* fku   = (const float*)d_in[0];   // feat_k_user   (R,N_USER,D)
    const float* fkm   = (const float*)d_in[1];   // feat_k_movie  (R,N_MOVIE,D)
    const float* fsu   = (const float*)d_in[2];   // feat_sum_user (R,N_USER,F,D)
    const float* fsm   = (const float*)d_in[3];   // feat_sum_movie(R,N_MOVIE,F,D)
    const float* rfa   = (const float*)d_in[4];   // review_feat_all (R,E,F,RD)
    const float* proto = (const float*)d_in[5];   // prototypes (F,RD)
    const float* eta   = (const float*)d_in[6];   // eta (R,E)
    const float* nwf   = (const float*)d_in[7];   // node_w_fwd (R,D,D)
    const float* rwf   = (const float*)d_in[8];   // review_w_fwd (R,RD,D)
    const float* nwr   = (const float*)d_in[9];   // node_w_rev (R,D,D)
    const float* rwr   = (const float*)d_in[10];  // review_w_rev (R,RD,D)
    const float* ufcw  = (const float*)d_in[11];  // ufc_w (D,OUT)
    const float* ufcb  = (const float*)d_in[12];  // ufc_b (OUT)
    const float* ifcw  = (const float*)d_in[13];  // ifc_w (D,OUT)
    const float* ifcb  = (const float*)d_in[14];  // ifc_b (OUT)
    const int*   erow  = (const int*)d_in[15];    // edges_row (R,E)
    const int*   ecol  = (const int*)d_in[16];    // edges_col (R,E)

    // workspace layout (floats); accumulator block is contiguous at the front
    float* ws         = (float*)d_ws;
    float* norm_user  = ws;                                   // 100000
    float* norm_movie = norm_user + N_USER;                   // 50000
    float* ufeat_pre  = norm_movie + N_MOVIE;                 // 3.2M
    float* ifeat_pre  = ufeat_pre + (long)N_USER * D_DIM;     // 1.6M
    float* w_raw      = ifeat_pre + (long)N_MOVIE * D_DIM;    // 500K
    float* src_u      = w_raw + (long)R_REL * E_EDGE;         // 16M
    float* src_m      = src_u + (long)R_REL * N_USER * D_DIM; // 8M

    float* out_ufeat = (float*)d_out;                              // N_USER*OUT
    float* out_ifeat = out_ufeat + (long)N_USER * OUT_DIM;         // N_MOVIE*OUT
    float* out_int   = out_ifeat + (long)N_MOVIE * OUT_DIM;        // R*E

    const long n_zero = (long)N_USER + N_MOVIE
                      + (long)N_USER * D_DIM + (long)N_MOVIE * D_DIM;  // 4,950,000 (mult of 4? 4950000/4 = 1237500 exact)
    zero_kernel<<<dim3(1024), dim3(1024), 0, stream>>>((float4*)ws, n_zero / 4);

    // src_u[r] = feat_k_user[r] @ node_w_fwd[r]   (100000x32x32, 5 relations)
    wmma_gemm<0, 2><<<dim3((N_USER / 16 + 7) / 8, 1, R_REL), dim3(256), 0, stream>>>(
        fku, nwf, nullptr, src_u, N_USER, D_DIM,
        (long)N_USER * D_DIM, (long)D_DIM * D_DIM, (long)N_USER * D_DIM);

    // src_m[r] = feat_k_movie[r] @ node_w_rev[r]  (50000x32x32, 5 relations)
    wmma_gemm<0, 2><<<dim3((N_MOVIE / 16 + 7) / 8, 1, R_REL), dim3(256), 0, stream>>>(
        fkm, nwr, nullptr, src_m, N_MOVIE, D_DIM,
        (long)N_MOVIE * D_DIM, (long)D_DIM * D_DIM, (long)N_MOVIE * D_DIM);

    // per-edge weights + degree sums
    edge_weights<<<dim3((E_EDGE + 255) / 256, R_REL), dim3(256), 0, stream>>>(
        fku, fkm, fsu, fsm, rfa, proto, eta, erow, ecol,
        w_raw, norm_user, norm_movie);

    // per-edge messages + int_dist
    edge_messages<<<dim3((E_EDGE + 255) / 256, R_REL), dim3(256), 0, stream>>>(
        src_u, src_m, rfa, rwf, rwr, erow, ecol,
        w_raw, norm_user, norm_movie, ufeat_pre, ifeat_pre, out_int);

    // ufeat = leaky(ufeat_pre) @ ufc_w + ufc_b   (100000x32x64)
    wmma_gemm<1, 4><<<dim3((N_USER / 16 + 7) / 8, 1, 1), dim3(256), 0, stream>>>(
        ufeat_pre, ufcw, ufcb, out_ufeat, N_USER, D_DIM, 0, 0, 0);

    // ifeat = leaky(ifeat_pre) @ ifc_w + ifc_b   (50000x32x64)
    wmma_gemm<1, 4><<<dim3((N_MOVIE / 16 + 7) / 8, 1, 1), dim3(256), 0, stream>>>(
        ifeat_pre, ifcw, ifcb, out_ifeat, N_MOVIE, D_DIM, 0, 0, 0);
}